// GraphLam_Model_49555332662124
// MI455X (gfx1250) — compile-verified
//
#include <hip/hip_runtime.h>

// GraphLam forward, MI455X (gfx1250, wave32, WMMA).
// Live dataflow only (the edge gather/scatter path in _inet is dead code):
//   rows [0,NG)        : OutMLP(((MLPg(x) @ Wa) @ Wb) @ Wc)
//   rows [NG,NG+NM)    : OutMLP(((MLPm(x) @ Wa) @ Wb) @ Wc)
//   rows [NG+NM,2NG+NM): OutMLP(MLPenc(MLPg(x)) @ Wc)
// All GEMMs are 16x32xK tiles on V_WMMA_F32_16X16X4_F32 (fp32, exact dtype).

typedef float v2f __attribute__((ext_vector_type(2)));
typedef float v8f __attribute__((ext_vector_type(8)));

#define NGRID  262144
#define NMESH  65536
#define ODIM   17

// packed-weight offsets (in floats) inside d_ws
#define PK_WG1    0        // grid.W1  52x32 (13 kslices) -> 1664 floats
#define PK_WG2    1664     // grid.W2  32x32
#define PK_WM1    2688     // mesh.W1  3x32 padded to 4x32 -> 128
#define PK_WM2    2816     // mesh.W2
#define PK_WE1    3840     // enc.W1
#define PK_WE2    4864     // enc.W2
#define PK_WA     5888     // g2m_gnn.Wrn
#define PK_WB     6912     // proc.Wrn
#define PK_WC     7936     // m2g_gnn.Wrn
#define PK_WO1    8960     // out.Wo1
#define PK_WO2    9984     // out.Wo2 32x17 padded to 32x32
#define PK_TOTAL  11008    // ~43 KB of workspace

// ---------------------------------------------------------------------------
// Weight packing: B-fragment layout per CDNA5 ISA 7.12.2 (32-bit B 4x16):
//   lanes 0-15 : VGPR0=K(4ks+0), VGPR1=K(4ks+1), N=lane
//   lanes 16-31: VGPR0=K(4ks+2), VGPR1=K(4ks+3), N=lane-16
// packed[((t*KS+ks)*32+lane)*2 + j] = W[4ks + 2*(lane>=16) + j][16t + (lane&15)]
// ---------------------------------------------------------------------------
__global__ void pack_weights_kernel(
    const float* __restrict__ Wg1, const float* __restrict__ Wg2,
    const float* __restrict__ Wm1, const float* __restrict__ Wm2,
    const float* __restrict__ We1, const float* __restrict__ We2,
    const float* __restrict__ Wa,  const float* __restrict__ Wb,
    const float* __restrict__ Wc,  const float* __restrict__ Wo1,
    const float* __restrict__ Wo2, float* __restrict__ pk)
{
  const float* srcs[11] = {Wg1, Wg2, Wm1, Wm2, We1, We2, Wa, Wb, Wc, Wo1, Wo2};
  const int ksv[11]  = {13, 8, 1, 8, 8, 8, 8, 8, 8, 8, 8};
  const int rkv[11]  = {52, 32, 3, 32, 32, 32, 32, 32, 32, 32, 32};
  const int rnv[11]  = {32, 32, 32, 32, 32, 32, 32, 32, 32, 32, 17};
  const int offv[11] = {PK_WG1, PK_WG2, PK_WM1, PK_WM2, PK_WE1, PK_WE2,
                        PK_WA, PK_WB, PK_WC, PK_WO1, PK_WO2};
  const int tid = threadIdx.x;
  for (int m = 0; m < 11; ++m) {
    const int KS = ksv[m], realK = rkv[m], realN = rnv[m];
    const int total = KS * 128;                 // 2 tiles * KS * 32 lanes * 2
    const float* __restrict__ src = srcs[m];
    for (int p = tid; p < total; p += blockDim.x) {
      const int j    = p & 1;
      const int q    = p >> 1;
      const int lane = q & 31;
      const int u    = q >> 5;                  // [0, 2*KS)
      const int ks   = u % KS;
      const int t    = u / KS;
      const int k    = 4 * ks + ((lane >> 4) & 1) * 2 + j;
      const int n    = 16 * t + (lane & 15);
      const float v  = (k < realK && n < realN) ? src[k * realN + n] : 0.0f;
      pk[offv[m] + p] = v;
    }
  }
}

// ---------------------------------------------------------------------------
// Wave-level 16-row GEMM/elementwise primitives
// ---------------------------------------------------------------------------
__device__ __forceinline__ v8f wmma4(v2f a, v2f b, v8f c) {
  // D = A(16x4,f32) x B(4x16,f32) + C(16x16,f32)
  return __builtin_amdgcn_wmma_f32_16x16x4_f32(
      false, a, false, b, (short)0, c, false, false);
}

struct Tile { v8f t0, t1; };   // 16x32 result: two 16x16 D tiles

// A: 16 rows x (4*KS) cols, row stride lda floats (global or LDS).
// pk: packed B for a (4*KS)x32 weight.
template <int KS>
__device__ __forceinline__ Tile gemm16(const float* __restrict__ A, int lda,
                                       const float* __restrict__ pk)
{
  const int lane = threadIdx.x & 31;
  const int row  = lane & 15;
  const int koff = ((lane >> 4) & 1) * 2;   // A layout: upper half-wave gets K+2
  v8f c0 = {0.f, 0.f, 0.f, 0.f, 0.f, 0.f, 0.f, 0.f};
  v8f c1 = c0;
#pragma unroll
  for (int ks = 0; ks < KS; ++ks) {
    const v2f a  = *(const v2f*)(A + row * lda + 4 * ks + koff);
    const v2f b0 = *(const v2f*)(pk + (ks * 32 + lane) * 2);
    const v2f b1 = *(const v2f*)(pk + ((KS + ks) * 32 + lane) * 2);
    c0 = wmma4(a, b0, c0);
    c1 = wmma4(a, b1, c1);
  }
  Tile r; r.t0 = c0; r.t1 = c1; return r;
}

// First mesh GEMM: X is 16x3 row-major, K padded 3->4 (zero lane component).
__device__ __forceinline__ Tile gemm_mesh(const float* __restrict__ X,
                                          const float* __restrict__ pk)
{
  const int lane = threadIdx.x & 31;
  const int row  = lane & 15;
  const int k0   = ((lane >> 4) & 1) * 2;   // 0 or 2
  v2f a;
  a.x = X[row * 3 + k0];
  a.y = (k0 + 1 < 3) ? X[row * 3 + k0 + 1] : 0.0f;
  const v2f b0 = *(const v2f*)(pk + lane * 2);
  const v2f b1 = *(const v2f*)(pk + (32 + lane) * 2);
  v8f z = {0.f, 0.f, 0.f, 0.f, 0.f, 0.f, 0.f, 0.f};
  Tile r; r.t0 = wmma4(a, b0, z); r.t1 = wmma4(a, b1, z); return r;
}

__device__ __forceinline__ void add_bias(Tile& T, const float* __restrict__ bias) {
  const int c = threadIdx.x & 15;
  const float s0 = bias[c], s1 = bias[c + 16];
#pragma unroll
  for (int r = 0; r < 8; ++r) { T.t0[r] += s0; T.t1[r] += s1; }
}

__device__ __forceinline__ void add_bias_out(Tile& T, const float* __restrict__ bias) {
  const int c = threadIdx.x & 15;
  const float s0 = bias[c];                       // cols 0..15 (all < 17)
  const float s1 = (c == 0) ? bias[16] : 0.0f;    // col 16 only; 17..31 unused
#pragma unroll
  for (int r = 0; r < 8; ++r) { T.t0[r] += s0; T.t1[r] += s1; }
}

__device__ __forceinline__ void silu_t(Tile& T) {
#pragma unroll
  for (int r = 0; r < 8; ++r) {
    T.t0[r] = T.t0[r] / (1.0f + __expf(-T.t0[r]));
    T.t1[r] = T.t1[r] / (1.0f + __expf(-T.t1[r]));
  }
}

// LayerNorm over the 32 features of each row. In D-tile layout row M's 16
// values per tile live in one VGPR across one 16-lane half, so xor-shuffles
// with masks 1..8 reduce exactly over that row.
__device__ __forceinline__ void layer_norm(Tile& T, const float* __restrict__ g,
                                           const float* __restrict__ b)
{
  const int c = threadIdx.x & 15;
  const float g0 = g[c], g1 = g[c + 16], bb0 = b[c], bb1 = b[c + 16];
#pragma unroll
  for (int r = 0; r < 8; ++r) {
    float s = T.t0[r] + T.t1[r];
    float q = T.t0[r] * T.t0[r] + T.t1[r] * T.t1[r];
#pragma unroll
    for (int m = 1; m < 16; m <<= 1) {
      s += __shfl_xor(s, m, 32);
      q += __shfl_xor(q, m, 32);
    }
    const float mean = s * (1.0f / 32.0f);
    const float var  = q * (1.0f / 32.0f) - mean * mean;
    const float rstd = rsqrtf(var + 1e-5f);
    T.t0[r] = (T.t0[r] - mean) * rstd * g0 + bb0;
    T.t1[r] = (T.t1[r] - mean) * rstd * g1 + bb1;
  }
}

// D-tile (C layout) -> LDS as 16x32 row-major (becomes next GEMM's A).
__device__ __forceinline__ void stage(const Tile& T, float* S) {
  const int lane = threadIdx.x & 31;
  const int c    = lane & 15;
  const int half = (lane >> 4) & 1;
#pragma unroll
  for (int r = 0; r < 8; ++r) {
    S[(r + 8 * half) * 32 + c]      = T.t0[r];
    S[(r + 8 * half) * 32 + c + 16] = T.t1[r];
  }
}

__device__ __forceinline__ void store_out(const Tile& T, float* __restrict__ out,
                                          int rowBase)
{
  const int lane = threadIdx.x & 31;
  const int c    = lane & 15;
  const int half = (lane >> 4) & 1;
#pragma unroll
  for (int r = 0; r < 8; ++r) {
    const int row = rowBase + r + 8 * half;
    out[row * ODIM + c] = T.t0[r];
    if (c == 0) out[row * ODIM + 16] = T.t1[r];
  }
}

// output_map: silu(x@Wo1+bo1)@Wo2+bo2 on the 16x32 tile in S, store 17 cols.
__device__ __forceinline__ void out_head(float* S, const float* __restrict__ pk,
                                         const float* __restrict__ bo1,
                                         const float* __restrict__ bo2,
                                         float* __restrict__ out, int rowBase)
{
  Tile T = gemm16<8>(S, 32, pk + PK_WO1);
  add_bias(T, bo1);
  silu_t(T);
  stage(T, S);
  T = gemm16<8>(S, 32, pk + PK_WO2);
  add_bias_out(T, bo2);
  store_out(T, out, rowBase);
}

// ---------------------------------------------------------------------------
// Grid-node kernel: each wave fully processes 16 grid rows, writing output
// rows [g0,g0+16) (chain path) and [NG+NM+g0, ...) (encoder path).
// ---------------------------------------------------------------------------
__global__ __launch_bounds__(256) void grid_path_kernel(
    const float* __restrict__ gridf, const float* __restrict__ pk,
    const float* __restrict__ gb1, const float* __restrict__ gb2,
    const float* __restrict__ gg,  const float* __restrict__ gbt,
    const float* __restrict__ eb1, const float* __restrict__ eb2,
    const float* __restrict__ eg,  const float* __restrict__ ebt,
    const float* __restrict__ bo1, const float* __restrict__ bo2,
    float* __restrict__ out)
{
  __shared__ float smem[8 * 1024];            // 2 staging tiles per wave
  const int wave = threadIdx.x >> 5;
  float* S  = smem + wave * 1024;             // holds grid_emb
  float* S2 = S + 512;                        // scratch tile
  const int g0 = (blockIdx.x * 8 + wave) * 16;

  // grid_emb = LN(silu(X@W1+b1)@W2+b2)*g+b
  Tile T = gemm16<13>(gridf + g0 * 52, 52, pk + PK_WG1);
  add_bias(T, gb1); silu_t(T); stage(T, S2);
  T = gemm16<8>(S2, 32, pk + PK_WG2);
  add_bias(T, gb2); layer_norm(T, gg, gbt);
  stage(T, S);                                // S = grid_emb (kept)

  // rows [g0, g0+16): ((e @ Wa) @ Wb) @ Wc -> out head
  T = gemm16<8>(S, 32, pk + PK_WA);  stage(T, S2);
  T = gemm16<8>(S2, 32, pk + PK_WB); stage(T, S2);
  T = gemm16<8>(S2, 32, pk + PK_WC); stage(T, S2);
  out_head(S2, pk, bo1, bo2, out, g0);

  // rows [NG+NM+g0, ...): MLP_enc(e) @ Wc -> out head
  T = gemm16<8>(S, 32, pk + PK_WE1);
  add_bias(T, eb1); silu_t(T); stage(T, S2);
  T = gemm16<8>(S2, 32, pk + PK_WE2);
  add_bias(T, eb2); layer_norm(T, eg, ebt); stage(T, S2);
  T = gemm16<8>(S2, 32, pk + PK_WC); stage(T, S2);
  out_head(S2, pk, bo1, bo2, out, NGRID + NMESH + g0);
}

// ---------------------------------------------------------------------------
// Mesh-node kernel: 16 mesh rows per wave, writes output rows [NG+m0, ...).
// ---------------------------------------------------------------------------
__global__ __launch_bounds__(256) void mesh_path_kernel(
    const float* __restrict__ meshf, const float* __restrict__ pk,
    const float* __restrict__ mb1, const float* __restrict__ mb2,
    const float* __restrict__ mg,  const float* __restrict__ mbt,
    const float* __restrict__ bo1, const float* __restrict__ bo2,
    float* __restrict__ out)
{
  __shared__ float smem[8 * 512];
  const int wave = threadIdx.x >> 5;
  float* S = smem + wave * 512;
  const int m0 = (blockIdx.x * 8 + wave) * 16;

  Tile T = gemm_mesh(meshf + m0 * 3, pk + PK_WM1);
  add_bias(T, mb1); silu_t(T); stage(T, S);
  T = gemm16<8>(S, 32, pk + PK_WM2);
  add_bias(T, mb2); layer_norm(T, mg, mbt); stage(T, S);
  T = gemm16<8>(S, 32, pk + PK_WA); stage(T, S);
  T = gemm16<8>(S, 32, pk + PK_WB); stage(T, S);
  T = gemm16<8>(S, 32, pk + PK_WC); stage(T, S);
  out_head(S, pk, bo1, bo2, out, NGRID + m0);
}

// ---------------------------------------------------------------------------
// Launch. Input indexing assumes recursive insertion-order flattening of
// setup_inputs(): 8 top-level tensors, then params leaves:
//   8..13  grid  {W1,b1,W2,b2,g,b}     14..19 mesh {...}
//   20..25 g2m (dead)  26..31 m2g (dead)  32..37 m2m (dead)
//   38..43 enc {...}
//   44..49 g2m_gnn {Ws,We,Wr,Weu,Wrn,Wagg}  -> Wrn = 48
//   50..55 proc                             -> Wrn = 54
//   56..61 m2g_gnn                          -> Wrn = 60
//   62..65 out {Wo1,bo1,Wo2,bo2}
// ---------------------------------------------------------------------------
extern "C" void kernel_launch(void* const* d_in, const int* in_sizes, int n_in,
                              void* d_out, int out_size, void* d_ws, size_t ws_size,
                              hipStream_t stream)
{
  (void)in_sizes; (void)n_in; (void)out_size; (void)ws_size;
  const float* gridf = (const float*)d_in[2];
  const float* meshf = (const float*)d_in[6];

  const float* gW1 = (const float*)d_in[8];
  const float* gb1 = (const float*)d_in[9];
  const float* gW2 = (const float*)d_in[10];
  const float* gb2 = (const float*)d_in[11];
  const float* gg  = (const float*)d_in[12];
  const float* gbt = (const float*)d_in[13];

  const float* mW1 = (const float*)d_in[14];
  const float* mb1 = (const float*)d_in[15];
  const float* mW2 = (const float*)d_in[16];
  const float* mb2 = (const float*)d_in[17];
  const float* mg  = (const float*)d_in[18];
  const float* mbt = (const float*)d_in[19];

  const float* eW1 = (const float*)d_in[38];
  const float* eb1 = (const float*)d_in[39];
  const float* eW2 = (const float*)d_in[40];
  const float* eb2 = (const float*)d_in[41];
  const float* eg  = (const float*)d_in[42];
  const float* ebt = (const float*)d_in[43];

  const float* Wa  = (const float*)d_in[48];   // g2m_gnn.Wrn
  const float* Wb  = (const float*)d_in[54];   // proc.Wrn
  const float* Wc  = (const float*)d_in[60];   // m2g_gnn.Wrn

  const float* Wo1 = (const float*)d_in[62];
  const float* bo1 = (const float*)d_in[63];
  const float* Wo2 = (const float*)d_in[64];
  const float* bo2 = (const float*)d_in[65];

  float* pk  = (float*)d_ws;
  float* out = (float*)d_out;

  pack_weights_kernel<<<1, 1024, 0, stream>>>(
      gW1, gW2, mW1, mW2, eW1, eW2, Wa, Wb, Wc, Wo1, Wo2, pk);

  grid_path_kernel<<<NGRID / 16 / 8, 256, 0, stream>>>(
      gridf, pk, gb1, gb2, gg, gbt, eb1, eb2, eg, ebt, bo1, bo2, out);

  mesh_path_kernel<<<NMESH / 16 / 8, 256, 0, stream>>>(
      meshf, pk, mb1, mb2, mg, mbt, bo1, bo2, out);
}